// GaussianKernelLoss_14748917694741
// MI455X (gfx1250) — compile-verified
//
#include <hip/hip_runtime.h>

#define HH 256
#define WW 256
#define RADIUS_F 0.2f
#define R2_F 0.04f
#define INV_2SIG2 50.0f   // 1/(2*0.1^2)
#define NLAND 512         // B*N = 32*16

typedef __attribute__((ext_vector_type(2))) float v2f;
typedef __attribute__((ext_vector_type(8))) float v8f;

// One block per landmark. Iterate only the bounding box around the target
// (hard mask at RADIUS makes everything outside contribute exactly 0).
__global__ __launch_bounds__(256)
void gauss_landmark_partial(const float* __restrict__ pred,
                            const float* __restrict__ tgt,
                            const float* __restrict__ vis,
                            const float* __restrict__ grid,  // [2][H][W]
                            float* __restrict__ ws)          // [NLAND]
{
    const int l   = blockIdx.x;      // landmark index (b*N + n)
    const int tid = threadIdx.x;

    const float tx = tgt[2 * l + 0];
    const float ty = tgt[2 * l + 1];
    const float px = pred[2 * l + 0];
    const float py = pred[2 * l + 1];

    // Conservative pixel bounding box (linspace step = 1/255), padded 1 px.
    int j0 = (int)floorf((tx - RADIUS_F) * 255.0f) - 1; if (j0 < 0) j0 = 0;
    int j1 = (int)ceilf ((tx + RADIUS_F) * 255.0f) + 1; if (j1 > WW - 1) j1 = WW - 1;
    int i0 = (int)floorf((ty - RADIUS_F) * 255.0f) - 1; if (i0 < 0) i0 = 0;
    int i1 = (int)ceilf ((ty + RADIUS_F) * 255.0f) + 1; if (i1 > HH - 1) i1 = HH - 1;

    const int bw    = j1 - j0 + 1;
    const int total = bw * (i1 - i0 + 1);

    const float* __restrict__ gx = grid;            // channel 0 = x
    const float* __restrict__ gy = grid + HH * WW;  // channel 1 = y

    float num = 0.0f, den = 0.0f;
    for (int idx = tid; idx < total; idx += 256) {
        const int r = idx / bw;
        const int i = i0 + r;
        const int j = j0 + (idx - r * bw);
        const int o = i * WW + j;
        const float x = gx[o];
        const float y = gy[o];
        const float dxt = x - tx, dyt = y - ty;
        const float dt2 = dxt * dxt + dyt * dyt;
        if (dt2 <= R2_F) {
            const float w   = __expf(-dt2 * INV_2SIG2);
            const float dxp = x - px, dyp = y - py;
            const float dp  = sqrtf(dxp * dxp + dyp * dyp);
            num = fmaf(w, dp, num);
            den += w;
        }
    }

    // Intra-wave (wave32) reduction.
    for (int off = 16; off > 0; off >>= 1) {
        num += __shfl_down(num, off, 32);
        den += __shfl_down(den, off, 32);
    }

    // Cross-wave reduction through LDS (8 waves per block).
    __shared__ float snum[8];
    __shared__ float sden[8];
    if ((tid & 31) == 0) { snum[tid >> 5] = num; sden[tid >> 5] = den; }
    __syncthreads();
    if (tid == 0) {
        float n = 0.0f, d = 0.0f;
        #pragma unroll
        for (int w = 0; w < 8; ++w) { n += snum[w]; d += sden[w]; }
        ws[l] = (vis[l] >= 0.5f) ? (n / (d + 1e-8f)) : 0.0f;
    }
}

// Final exact-f32 reduction of the 512 per-landmark values using the matrix
// pipe: D = A x ones + C with V_WMMA_F32_16X16X4_F32. Every column of D holds
// the 16 row sums, so summing one lane's 8 accumulator VGPRs plus lane^16's
// gives the full 64-element tile total independent of A's lane->(m,k) layout.
__global__ __launch_bounds__(32)
void reduce512_wmma(const float* __restrict__ ws, float* __restrict__ out)
{
    const int lane = threadIdx.x;         // 0..31, EXEC all ones
    const int m  = lane & 15;
    const int kb = (lane >> 4) * 2;

    v2f b; b.x = 1.0f; b.y = 1.0f;        // ones B-matrix (4x16 f32)
    v8f acc = {};                          // C/D accumulator (16x16 f32)

    #pragma unroll
    for (int t = 0; t < 8; ++t) {         // 8 tiles x 64 values = 512
        const int base = t * 64 + m * 4 + kb;
        v2f a; a.x = ws[base]; a.y = ws[base + 1];
        acc = __builtin_amdgcn_wmma_f32_16x16x4_f32(
            /*neg_a=*/false, a, /*neg_b=*/false, b,
            /*c_mod=*/(short)0, acc, /*reuse_a=*/false, /*reuse_b=*/false);
    }

    float s = 0.0f;
    #pragma unroll
    for (int r = 0; r < 8; ++r) s += acc[r];       // half-column sum
    const float other = __shfl(s, lane ^ 16, 32);  // partner half-column
    if (lane == 0) out[0] = (s + other) * (1.0f / 512.0f);
}

extern "C" void kernel_launch(void* const* d_in, const int* in_sizes, int n_in,
                              void* d_out, int out_size, void* d_ws, size_t ws_size,
                              hipStream_t stream) {
    (void)in_sizes; (void)n_in; (void)out_size; (void)ws_size;
    const float* pred = (const float*)d_in[0];  // [B,N,2]
    const float* tgt  = (const float*)d_in[1];  // [B,N,2]
    const float* vis  = (const float*)d_in[2];  // [B,N]
    const float* grid = (const float*)d_in[3];  // [2,H,W]
    float* out = (float*)d_out;                 // scalar
    float* ws  = (float*)d_ws;                  // >= 512 floats

    gauss_landmark_partial<<<NLAND, 256, 0, stream>>>(pred, tgt, vis, grid, ws);
    reduce512_wmma<<<1, 32, 0, stream>>>(ws, out);
}